// HashEncoderNative_3530463117846
// MI455X (gfx1250) — compile-verified
//
#include <hip/hip_runtime.h>
#include <hip/hip_bf16.h>
#include <stdint.h>

// Instant-NGP style 4D hash-grid encoder for MI455X (gfx1250).
// Memory-bound gather kernel: 64MB hash table is L2-resident (192MB L2),
// so we optimize for gather MLP (16 independent b64 loads in flight per
// thread) and coalesced b128/b64 streaming of coords and outputs.

#define NUM_LEVELS   16
#define LOG2_T       19
#define TABLE_SIZE   (1u << LOG2_T)
#define PTS_PER_BLOCK 16
#define BLOCK        256

// floor(16 * (256/16)^(i/15)) -- matches the reference's _SCALINGS table.
__constant__ float c_scale[NUM_LEVELS] = {
    16.f, 19.f, 23.f, 27.f, 33.f, 40.f, 48.f, 58.f,
    70.f, 84.f, 101.f, 122.f, 147.f, 176.f, 212.f, 256.f
};

#define PRIME1 2654435761u
#define PRIME2 805459861u
#define PRIME3 3674653429u

#if defined(__has_builtin)
#  if __has_builtin(__builtin_amdgcn_global_load_async_to_lds_b128)
#    define HAVE_ASYNC_LDS 1
#  endif
#endif
#ifndef HAVE_ASYNC_LDS
#  define HAVE_ASYNC_LDS 0
#endif

typedef int v4i __attribute__((vector_size(16)));
typedef __attribute__((address_space(1))) v4i glob_v4i_t;  // global b128
typedef __attribute__((address_space(3))) v4i lds_v4i_t;   // LDS b128

__global__ __launch_bounds__(BLOCK) void hash_encode_4d_kernel(
    const float4* __restrict__ xyzt,   // [N] packed (x,y,z,t)
    const float2* __restrict__ table,  // [L*T] packed (f0,f1)
    float2* __restrict__ out,          // [N*L] packed (f0,f1)
    int n_points)
{
    __shared__ __align__(16) float4 s_xyzt[PTS_PER_BLOCK];
    const int point_base = blockIdx.x * PTS_PER_BLOCK;

    // Stage this block's 16 points (256B) into LDS once instead of 16
    // redundant per-lane b128 reads. Use the CDNA5 async DMA path when
    // the toolchain exposes it.
#if HAVE_ASYNC_LDS
    if (threadIdx.x < PTS_PER_BLOCK) {
        int pn = point_base + (int)threadIdx.x;
        if (pn >= n_points) pn = n_points - 1;
        __builtin_amdgcn_global_load_async_to_lds_b128(
            (glob_v4i_t*)(xyzt + pn),
            (lds_v4i_t*)&s_xyzt[threadIdx.x],
            /*offset=*/0, /*cpol=*/0);
    }
    asm volatile("s_wait_asynccnt 0" ::: "memory");
    __syncthreads();
#else
    if (threadIdx.x < PTS_PER_BLOCK) {
        int pn = point_base + (int)threadIdx.x;
        if (pn >= n_points) pn = n_points - 1;
        s_xyzt[threadIdx.x] = xyzt[pn];
    }
    __syncthreads();
#endif

    const int gid = blockIdx.x * BLOCK + (int)threadIdx.x;
    const int n   = gid >> 4;                 // point index (level fastest)
    const int l   = gid & (NUM_LEVELS - 1);   // level index
    if (n >= n_points) return;

    const float4 p = s_xyzt[threadIdx.x >> 4];
    const float  s = c_scale[l];

    const float x = p.x * s, y = p.y * s, z = p.z * s, t = p.w * s;

    const float fx = floorf(x), fy = floorf(y), fz = floorf(z), ft = floorf(t);
    const float ox = x - fx,    oy = y - fy,    oz = z - fz,    ot = t - ft;

    // Per-dimension hash contributions (uint32 wrap-around multiply matches
    // the reference's int64-mul + mod-2^19 exactly since 2^19 | 2^32).
    const uint32_t hx0 = (uint32_t)fx;
    const uint32_t hy0 = (uint32_t)fy * PRIME1;
    const uint32_t hz0 = (uint32_t)fz * PRIME2;
    const uint32_t ht0 = (uint32_t)ft * PRIME3;
    const uint32_t hx1 = (uint32_t)ceilf(x);
    const uint32_t hy1 = (uint32_t)ceilf(y) * PRIME1;
    const uint32_t hz1 = (uint32_t)ceilf(z) * PRIME2;
    const uint32_t ht1 = (uint32_t)ceilf(t) * PRIME3;

    const uint32_t base = (uint32_t)l << LOG2_T;

    const float wx1 = ox, wx0 = 1.0f - ox;
    const float wy1 = oy, wy0 = 1.0f - oy;
    const float wz1 = oz, wz0 = 1.0f - oz;
    const float wt1 = ot, wt0 = 1.0f - ot;

    // Phase 1: issue all 16 independent gathers (global_load_b64) so the
    // wave has maximal loads in flight against the L2-resident table.
    float2 feat[16];
#pragma unroll
    for (int c = 0; c < 16; ++c) {
        const uint32_t h = ((c & 1) ? hx1 : hx0)
                         ^ ((c & 2) ? hy1 : hy0)
                         ^ ((c & 4) ? hz1 : hz0)
                         ^ ((c & 8) ? ht1 : ht0);
        feat[c] = table[(h & (TABLE_SIZE - 1u)) + base];
    }

    // Phase 2: quadrilinear accumulation.
    float accx = 0.0f, accy = 0.0f;
#pragma unroll
    for (int c = 0; c < 16; ++c) {
        const float w = ((c & 1) ? wx1 : wx0)
                      * ((c & 2) ? wy1 : wy0)
                      * ((c & 4) ? wz1 : wz0)
                      * ((c & 8) ? wt1 : wt0);
        accx = fmaf(w, feat[c].x, accx);
        accy = fmaf(w, feat[c].y, accy);
    }

    out[gid] = make_float2(accx, accy);  // coalesced global_store_b64
}

extern "C" void kernel_launch(void* const* d_in, const int* in_sizes, int n_in,
                              void* d_out, int out_size, void* d_ws, size_t ws_size,
                              hipStream_t stream) {
    const float4* xyzt  = (const float4*)d_in[0];   // [N,4] float32
    const float2* table = (const float2*)d_in[1];   // [L*T,2] float32
    float2*       out   = (float2*)d_out;           // [N, L*2] float32

    const int n_points = in_sizes[0] / 4;
    const int total    = n_points * NUM_LEVELS;
    const int blocks   = (total + BLOCK - 1) / BLOCK;

    hipLaunchKernelGGL(hash_encode_4d_kernel,
                       dim3(blocks), dim3(BLOCK), 0, stream,
                       xyzt, table, out, n_points);
}